// BasisJastrow_12429635355198
// MI455X (gfx1250) — compile-verified
//
#include <hip/hip_runtime.h>
#include <hip/hip_bf16.h>

typedef __attribute__((ext_vector_type(2))) float v2f;
typedef __attribute__((ext_vector_type(8))) float v8f;

#define BATCH   4096
#define NPART   16
#define KDIM    35      // number of (n0,n1,n2) modes with sum<=4
#define KPAD    48      // pad K to 3 WMMA N-tiles / 9 K-steps of 4
#define KSPLIT  18      // lanes 0-15 compute k<18, lanes 16-31 compute k>=18
#define NPAIR   120     // 16*15/2
#define WPB     8       // waves per block (wave32)

// ns table: itertools.product(range(5),repeat=3), sum<=4, stable-sorted by sum
static constexpr int NS0[KDIM] = {0, 0,0,1, 0,0,0,1,1,2, 0,0,0,0,1,1,1,2,2,3,
                                  0,0,0,0,0,1,1,1,1,2,2,2,3,3,4};
static constexpr int NS1[KDIM] = {0, 0,1,0, 0,1,2,0,1,0, 0,1,2,3,0,1,2,0,1,0,
                                  0,1,2,3,4,0,1,2,3,0,1,2,0,1,0};
static constexpr int NS2[KDIM] = {0, 1,0,0, 2,1,0,1,0,0, 3,2,1,0,2,1,0,1,0,0,
                                  4,3,2,1,0,3,2,1,0,2,1,0,1,0,0};

__global__ __launch_bounds__(256)
void basis_jastrow_kernel(const float* __restrict__ ri,        // [B][16][3]
                          const float* __restrict__ x_scaled,  // [B][16][3]
                          const float* __restrict__ sigma,     // [35]
                          const float* __restrict__ C,         // [35*35]
                          float* __restrict__ out)             // [B]
{
    __shared__ float cpad[KPAD * KPAD];                 // padded Jastrow C (9 KB)
    __shared__ float basis_s [WPB][NPART][KPAD];        // 24 KB
    __shared__ float prefix_s[WPB][NPART][KPAD];        // 24 KB

    const int tid  = threadIdx.x;
    const int wave = tid >> 5;
    const int lane = tid & 31;
    const int b    = blockIdx.x * WPB + wave;

    // ---- stage zero-padded C into LDS: 48*48 = 9*256, uniform per-thread ----
    #pragma unroll
    for (int it = 0; it < (KPAD * KPAD) / 256; ++it) {
        const int idx = it * 256 + tid;
        const int i = idx / KPAD, j = idx % KPAD;
        cpad[idx] = (i < KDIM && j < KDIM) ? C[i * KDIM + j] : 0.0f;
    }

    // ---- basis for particle v = lane&15; lane-halves split the k-range ----
    const int v    = lane & 15;
    const int half = lane >> 4;
    const float* rp = ri       + ((size_t)b * NPART + v) * 3;
    const float* xp = x_scaled + ((size_t)b * NPART + v) * 3;

    float h[3][5];
    float r2 = 0.0f;
    #pragma unroll
    for (int d = 0; d < 3; ++d) {
        const float x = xp[d];
        const float r = rp[d];
        r2 += r * r;
        h[d][0] = 1.0f;
        h[d][1] = 2.0f * x;
        #pragma unroll
        for (int n = 2; n <= 4; ++n)
            h[d][n] = 2.0f * x * h[d][n - 1] - 2.0f * (float)(n - 1) * h[d][n - 2];
    }
    const float tmul = -0.5f * r2;

    if (half == 0) {
        // lanes 0-15: k = 0 .. 17
        #pragma unroll
        for (int k = 0; k < KSPLIT; ++k) {
            basis_s[wave][v][k] = h[0][NS0[k]] * h[1][NS1[k]] * h[2][NS2[k]]
                                * expf(tmul * sigma[k]);
        }
    } else {
        // lanes 16-31: k = 18 .. 34, plus zero the N-pad columns 35..47
        #pragma unroll
        for (int k = KSPLIT; k < KDIM; ++k) {
            basis_s[wave][v][k] = h[0][NS0[k]] * h[1][NS1[k]] * h[2][NS2[k]]
                                * expf(tmul * sigma[k]);
        }
        #pragma unroll
        for (int k = KDIM; k < KPAD; ++k) basis_s[wave][v][k] = 0.0f;
    }
    __syncthreads();

    // ---- exclusive prefix over particles, one column per lane (+wrap) ----
    #pragma unroll
    for (int cc = 0; cc < 2; ++cc) {
        const int c = lane + cc * 32;
        if (c < KPAD) {
            float p = 0.0f;
            #pragma unroll
            for (int u = 0; u < NPART; ++u) {
                prefix_s[wave][u][c] = p;      // prefix_0 = 0 (exclusive)
                p += basis_s[wave][u][c];
            }
        }
    }
    __syncthreads();

    // ---- WMMA: U(16x48) = Prefix(16x36) * C^T(36x48), f32 16x16x4, 9 K-steps ----
    v8f acc0 = {0.f,0.f,0.f,0.f,0.f,0.f,0.f,0.f};
    v8f acc1 = {0.f,0.f,0.f,0.f,0.f,0.f,0.f,0.f};
    v8f acc2 = {0.f,0.f,0.f,0.f,0.f,0.f,0.f,0.f};
    const int m15 = lane & 15;        // A row (particle) / B column (mode)
    const int kh  = half * 2;         // per-lane K offset (ISA A/B layout)

    #pragma unroll
    for (int kk = 0; kk < 9; ++kk) {
        const int j0 = kk * 4;
        // A[m, k] = Prefix[b, m, j0+k]
        const v2f a  = *(const v2f*)&prefix_s[wave][m15][j0 + kh];
        // B[k, n] = C^T[j0+k, i0+n] = Cpad[i0+n, j0+k]
        const v2f b0 = *(const v2f*)&cpad[( 0 + m15) * KPAD + j0 + kh];
        const v2f b1 = *(const v2f*)&cpad[(16 + m15) * KPAD + j0 + kh];
        const v2f b2 = *(const v2f*)&cpad[(32 + m15) * KPAD + j0 + kh];
        acc0 = __builtin_amdgcn_wmma_f32_16x16x4_f32(false, a, false, b0,
                                                     (short)0, acc0, false, false);
        acc1 = __builtin_amdgcn_wmma_f32_16x16x4_f32(false, a, false, b1,
                                                     (short)0, acc1, false, false);
        acc2 = __builtin_amdgcn_wmma_f32_16x16x4_f32(false, a, false, b2,
                                                     (short)0, acc2, false, false);
    }

    // ---- J[b] = (1/P) * sum_{m,n} basis[b,m,n] * U[m,n]  (pads are zero) ----
    float s = 0.0f;
    #pragma unroll
    for (int r = 0; r < 8; ++r) {
        const int mm = r + half * 8;           // D-tile: VGPR r, lane-half -> M
        s += acc0[r] * basis_s[wave][mm][ 0 + m15];
        s += acc1[r] * basis_s[wave][mm][16 + m15];
        s += acc2[r] * basis_s[wave][mm][32 + m15];
    }
    #pragma unroll
    for (int off = 16; off > 0; off >>= 1)
        s += __shfl_xor(s, off, 32);
    if (lane == 0) out[b] = s * (1.0f / (float)NPAIR);
}

extern "C" void kernel_launch(void* const* d_in, const int* in_sizes, int n_in,
                              void* d_out, int out_size, void* d_ws, size_t ws_size,
                              hipStream_t stream) {
    (void)in_sizes; (void)n_in; (void)d_ws; (void)ws_size; (void)out_size;
    // inputs: 0=rij_dist (UNUSED by reference), 1=ri, 2=x_scaled, 3=sigma, 4=jastrow_coeff
    const float* ri    = (const float*)d_in[1];
    const float* xs    = (const float*)d_in[2];
    const float* sigma = (const float*)d_in[3];
    const float* C     = (const float*)d_in[4];
    float* out = (float*)d_out;

    dim3 grid(BATCH / WPB);   // 512 blocks * 8 waves = 4096 batch elements
    dim3 block(256);          // 8 wave32
    basis_jastrow_kernel<<<grid, block, 0, stream>>>(ri, xs, sigma, C, out);
}